// InteractionBlock_10634339024903
// MI455X (gfx1250) — compile-verified
//
#include <hip/hip_runtime.h>
#include <math.h>

// ---------------------------------------------------------------------------
// MACE-like interaction block for MI455X (gfx1250), fp32 end-to-end.
// All small GEMMs use V_WMMA_F32_16X16X4_F32 (exact fp32 matrix core path).
// Edge scatter uses global_atomic_add_f32 (accumulators fit in 192MB L2).
// ---------------------------------------------------------------------------

typedef float v2f __attribute__((ext_vector_type(2)));
typedef float v8f __attribute__((ext_vector_type(8)));

#define WMMA_F32(a, b, c) \
  __builtin_amdgcn_wmma_f32_16x16x4_f32(false, (a), false, (b), (short)0, (c), false, false)

#define MULC 32
#define INV_SQRT3F 0.57735026918962576f
#define INV_SQRT8F 0.35355339059327379f   // 1/sqrt(8)
#define L1F        0.17677669529663689f   // 1/sqrt(32)
#define L2F        0.125f                 // 1/sqrt(64)
#define SCNF       0.08838834764831845f   // 1/sqrt(128)

// ---------------------------------------------------------------------------
__global__ void zero_kernel(float* __restrict__ p, long n) {
  long i = (long)blockIdx.x * blockDim.x + threadIdx.x;
  long stride = (long)gridDim.x * blockDim.x;
  for (; i < n; i += stride) p[i] = 0.0f;
}

// ---------------------------------------------------------------------------
// Per-node pre-mix: s1 = s @ W1_s * l1, v1 = v @ W1_v * l1 (per component).
// One 16-node tile per wave; WMMA f32 16x16x4, K=32 (8 k-steps), N=32 (2 tiles).
__global__ __launch_bounds__(256) void node_pre_kernel(
    const float* __restrict__ nf, const float* __restrict__ W1s,
    const float* __restrict__ W1v, float* __restrict__ s1,
    float* __restrict__ v1, int N) {
  const int t = threadIdx.x;
  const int wv = t >> 5, lane = t & 31, hw = lane >> 4, lm = lane & 15;
  const long m0 = (long)blockIdx.x * 128 + wv * 16;
  if (m0 >= N) return;
  long mrow = m0 + lm;
  if (mrow >= N) mrow = N - 1;

  // ---- scalar channel: (16 x 32) @ (32 x 32)
  v8f c0 = {}, c1 = {};
  for (int ks = 0; ks < 8; ++ks) {
    const int kb = ks * 4 + 2 * hw;  // this lane's K pair base
    v2f a;
    a.x = nf[mrow * 128 + kb];
    a.y = nf[mrow * 128 + kb + 1];
    v2f b0, b1;
    b0.x = W1s[kb * 32 + lm];        b0.y = W1s[(kb + 1) * 32 + lm];
    b1.x = W1s[kb * 32 + 16 + lm];   b1.y = W1s[(kb + 1) * 32 + 16 + lm];
    c0 = WMMA_F32(a, b0, c0);
    c1 = WMMA_F32(a, b1, c1);
  }
  for (int j = 0; j < 8; ++j) {
    const long m = m0 + j + 8 * hw;
    if (m < N) {
      s1[m * 32 + lm]      = c0[j] * L1F;
      s1[m * 32 + 16 + lm] = c1[j] * L1F;
    }
  }

  // ---- vector channels (i = 0..2)
  for (int i = 0; i < 3; ++i) {
    v8f d0 = {}, d1 = {};
    for (int ks = 0; ks < 8; ++ks) {
      const int kb = ks * 4 + 2 * hw;
      v2f a;
      a.x = nf[mrow * 128 + 32 + kb * 3 + i];
      a.y = nf[mrow * 128 + 32 + (kb + 1) * 3 + i];
      v2f b0, b1;
      b0.x = W1v[kb * 32 + lm];       b0.y = W1v[(kb + 1) * 32 + lm];
      b1.x = W1v[kb * 32 + 16 + lm];  b1.y = W1v[(kb + 1) * 32 + 16 + lm];
      d0 = WMMA_F32(a, b0, d0);
      d1 = WMMA_F32(a, b1, d1);
    }
    for (int j = 0; j < 8; ++j) {
      const long m = m0 + j + 8 * hw;
      if (m < N) {
        v1[m * 96 + lm * 3 + i]        = d0[j] * L1F;
        v1[m * 96 + (16 + lm) * 3 + i] = d1[j] * L1F;
      }
    }
  }
}

// ---------------------------------------------------------------------------
// Edge kernel: per-wave 16-edge MLP via WMMA (staged through LDS), then
// block-wide gather / tensor-product / atomic scatter.
__global__ __launch_bounds__(256) void edge_kernel(
    const float* __restrict__ eattr, const float* __restrict__ eemb,
    const float* __restrict__ fcw1, const float* __restrict__ fcw2,
    const int* __restrict__ eidx, const float* __restrict__ s1,
    const float* __restrict__ v1, float* __restrict__ S,
    float* __restrict__ V, int E, int N) {
  __shared__ float lds_w[128 * 128];  // 64 KB: per-block w tile (128 edges x 128)
  const int t = threadIdx.x;
  const int wv = t >> 5, lane = t & 31, hw = lane >> 4, lm = lane & 15;
  const long eblk = (long)blockIdx.x * 128;
  const long e0 = eblk + wv * 16;

  // ---------------- phase 1: w = (silu(emb@fc_w1/sqrt8)) @ fc_w2 / sqrt8 ----
  {
    long me = e0 + lm;
    if (me >= E) me = E - 1;
    v2f a1[2], b1[2];
    for (int kk = 0; kk < 2; ++kk) {
      const int kb = kk * 4 + 2 * hw;
      a1[kk].x = eemb[me * 8 + kb];
      a1[kk].y = eemb[me * 8 + kb + 1];
      if (lm < 8) {  // fc_w1 is 8x8, zero-pad columns 8..15
        b1[kk].x = fcw1[kb * 8 + lm];
        b1[kk].y = fcw1[(kb + 1) * 8 + lm];
      } else {
        b1[kk].x = 0.0f; b1[kk].y = 0.0f;
      }
    }
    v8f c1 = {};
    c1 = WMMA_F32(a1[0], b1[0], c1);
    c1 = WMMA_F32(a1[1], b1[1], c1);

    // silu + stage h (transposed access next, so go through LDS)
    for (int j = 0; j < 8; ++j) {
      const float x = c1[j] * INV_SQRT8F;
      lds_w[(wv * 16 + j + 8 * hw) * 128 + lm] = x / (1.0f + __expf(-x));
    }
    // read h back in A-fragment layout (same-wave DS ordering handles RAW)
    v2f a2[2];
    for (int kk = 0; kk < 2; ++kk) {
      const int kb = kk * 4 + 2 * hw;
      a2[kk].x = lds_w[(wv * 16 + lm) * 128 + kb];
      a2[kk].y = lds_w[(wv * 16 + lm) * 128 + kb + 1];
    }
    for (int nt = 0; nt < 8; ++nt) {
      v2f b2[2];
      for (int kk = 0; kk < 2; ++kk) {
        const int kb = kk * 4 + 2 * hw;
        b2[kk].x = fcw2[kb * 128 + nt * 16 + lm];
        b2[kk].y = fcw2[(kb + 1) * 128 + nt * 16 + lm];
      }
      v8f c2 = {};
      c2 = WMMA_F32(a2[0], b2[0], c2);
      c2 = WMMA_F32(a2[1], b2[1], c2);
      for (int j = 0; j < 8; ++j)
        lds_w[(wv * 16 + j + 8 * hw) * 128 + nt * 16 + lm] = c2[j] * INV_SQRT8F;
    }
  }
  __syncthreads();

  // ---------------- phase 2: gather + tensor product + atomic scatter -------
  const int u = lane;  // channel index 0..31
  for (int it = 0; it < 16; ++it) {
    const int eL = (t >> 5) + 8 * it;  // 8 edges in flight per iteration
    const long e = eblk + eL;
    if (e < E) {
      const int dst = eidx[e];          // edge_index[0]
      const int src = eidx[(long)E + e];  // edge_index[1]
      const float a0  = eattr[e * 4 + 0];
      const float a1x = eattr[e * 4 + 1];
      const float a1y = eattr[e * 4 + 2];
      const float a1z = eattr[e * 4 + 3];
      const float ss  = s1[(long)src * 32 + u];
      const float vx  = v1[(long)src * 96 + u * 3 + 0];
      const float vy  = v1[(long)src * 96 + u * 3 + 1];
      const float vz  = v1[(long)src * 96 + u * 3 + 2];
      const float w0 = lds_w[eL * 128 + u];
      const float w1 = lds_w[eL * 128 + 32 + u];
      const float w2 = lds_w[eL * 128 + 64 + u];
      const float w3 = lds_w[eL * 128 + 96 + u];
      const float dot = vx * a1x + vy * a1y + vz * a1z;

      float* Sd = S + (long)dst * 64;
      float* Vd = V + (long)dst * 192;
      atomicAdd(&Sd[u],      w0 * ss * a0);
      atomicAdd(&Sd[32 + u], w3 * dot * INV_SQRT3F);
      atomicAdd(&Vd[u * 3 + 0], w1 * ss * a1x);
      atomicAdd(&Vd[u * 3 + 1], w1 * ss * a1y);
      atomicAdd(&Vd[u * 3 + 2], w1 * ss * a1z);
      atomicAdd(&Vd[96 + u * 3 + 0], w2 * vx * a0);
      atomicAdd(&Vd[96 + u * 3 + 1], w2 * vy * a0);
      atomicAdd(&Vd[96 + u * 3 + 2], w2 * vz * a0);
    }
  }
}

// ---------------------------------------------------------------------------
// Per-node output: s_out = S@W2_s*l2 + sc_s ; v_out = V@W2_v*l2 + sc_v.
// Self-connection folded into the same WMMA accumulator (K=64 + K=128 steps).
__global__ __launch_bounds__(256) void node_out_kernel(
    const float* __restrict__ nf, const float* __restrict__ nattr,
    const float* __restrict__ W2s, const float* __restrict__ W2v,
    const float* __restrict__ Wscs, const float* __restrict__ Wscv,
    const float* __restrict__ S, const float* __restrict__ V,
    float* __restrict__ out, int N) {
  const int t = threadIdx.x;
  const int wv = t >> 5, lane = t & 31, hw = lane >> 4, lm = lane & 15;
  const long m0 = (long)blockIdx.x * 128 + wv * 16;
  if (m0 >= N) return;
  long mrow = m0 + lm;
  if (mrow >= N) mrow = N - 1;

  // ---- scalar output
  v8f c0 = {}, c1 = {};
  for (int ks = 0; ks < 16; ++ks) {               // S(64) @ W2_s, scaled l2
    const int kb = ks * 4 + 2 * hw;
    v2f a;
    a.x = S[mrow * 64 + kb] * L2F;
    a.y = S[mrow * 64 + kb + 1] * L2F;
    v2f b0, b1;
    b0.x = W2s[kb * 32 + lm];       b0.y = W2s[(kb + 1) * 32 + lm];
    b1.x = W2s[kb * 32 + 16 + lm];  b1.y = W2s[(kb + 1) * 32 + 16 + lm];
    c0 = WMMA_F32(a, b0, c0);
    c1 = WMMA_F32(a, b1, c1);
  }
  for (int ks = 0; ks < 32; ++ks) {               // self-connection, K=128
    const int kb = ks * 4 + 2 * hw;
    v2f a;
    a.x = nf[mrow * 128 + (kb >> 2)] * nattr[mrow * 4 + (kb & 3)] * SCNF;
    a.y = nf[mrow * 128 + ((kb + 1) >> 2)] * nattr[mrow * 4 + ((kb + 1) & 3)] * SCNF;
    v2f b0, b1;
    b0.x = Wscs[kb * 32 + lm];       b0.y = Wscs[(kb + 1) * 32 + lm];
    b1.x = Wscs[kb * 32 + 16 + lm];  b1.y = Wscs[(kb + 1) * 32 + 16 + lm];
    c0 = WMMA_F32(a, b0, c0);
    c1 = WMMA_F32(a, b1, c1);
  }
  for (int j = 0; j < 8; ++j) {
    const long m = m0 + j + 8 * hw;
    if (m < N) {
      out[m * 128 + lm]      = c0[j];
      out[m * 128 + 16 + lm] = c1[j];
    }
  }

  // ---- vector outputs
  for (int i = 0; i < 3; ++i) {
    v8f d0 = {}, d1 = {};
    for (int ks = 0; ks < 16; ++ks) {
      const int kb = ks * 4 + 2 * hw;
      v2f a;
      a.x = V[mrow * 192 + kb * 3 + i] * L2F;
      a.y = V[mrow * 192 + (kb + 1) * 3 + i] * L2F;
      v2f b0, b1;
      b0.x = W2v[kb * 32 + lm];       b0.y = W2v[(kb + 1) * 32 + lm];
      b1.x = W2v[kb * 32 + 16 + lm];  b1.y = W2v[(kb + 1) * 32 + 16 + lm];
      d0 = WMMA_F32(a, b0, d0);
      d1 = WMMA_F32(a, b1, d1);
    }
    for (int ks = 0; ks < 32; ++ks) {
      const int kb = ks * 4 + 2 * hw;
      v2f a;
      a.x = nf[mrow * 128 + 32 + (kb >> 2) * 3 + i] * nattr[mrow * 4 + (kb & 3)] * SCNF;
      a.y = nf[mrow * 128 + 32 + ((kb + 1) >> 2) * 3 + i] * nattr[mrow * 4 + ((kb + 1) & 3)] * SCNF;
      v2f b0, b1;
      b0.x = Wscv[kb * 32 + lm];       b0.y = Wscv[(kb + 1) * 32 + lm];
      b1.x = Wscv[kb * 32 + 16 + lm];  b1.y = Wscv[(kb + 1) * 32 + 16 + lm];
      d0 = WMMA_F32(a, b0, d0);
      d1 = WMMA_F32(a, b1, d1);
    }
    for (int j = 0; j < 8; ++j) {
      const long m = m0 + j + 8 * hw;
      if (m < N) {
        out[m * 128 + 32 + lm * 3 + i]        = d0[j];
        out[m * 128 + 32 + (16 + lm) * 3 + i] = d1[j];
      }
    }
  }
}

// ---------------------------------------------------------------------------
extern "C" void kernel_launch(void* const* d_in, const int* in_sizes, int n_in,
                              void* d_out, int out_size, void* d_ws, size_t ws_size,
                              hipStream_t stream) {
  const float* nf    = (const float*)d_in[0];   // node_features (N,128)
  const float* nattr = (const float*)d_in[1];   // node_attrs    (N,4)
  const float* eattr = (const float*)d_in[2];   // edge_attrs    (E,4)
  const float* eemb  = (const float*)d_in[3];   // edge_embedding(E,8)
  const float* W1s   = (const float*)d_in[4];
  const float* W1v   = (const float*)d_in[5];
  const float* fcw1  = (const float*)d_in[6];
  const float* fcw2  = (const float*)d_in[7];
  const float* W2s   = (const float*)d_in[8];
  const float* W2v   = (const float*)d_in[9];
  const float* Wscs  = (const float*)d_in[10];
  const float* Wscv  = (const float*)d_in[11];
  const int*   eidx  = (const int*)d_in[12];    // edge_index (2,E) int32
  float* out = (float*)d_out;

  const int N = in_sizes[0] / 128;
  const int E = in_sizes[2] / 4;

  float* ws = (float*)d_ws;
  float* s1 = ws;                       // N*32
  float* v1 = ws + (long)N * 32;        // N*96
  float* S  = ws + (long)N * 128;       // N*64  (scatter accum, scalar)
  float* V  = ws + (long)N * 192;       // N*192 (scatter accum, vector)

  const long zn = (long)N * 256;        // S + V
  int zgrid = (int)((zn + 255) / 256);
  if (zgrid > 4096) zgrid = 4096;
  zero_kernel<<<zgrid, 256, 0, stream>>>(S, zn);

  node_pre_kernel<<<(N + 127) / 128, 256, 0, stream>>>(nf, W1s, W1v, s1, v1, N);
  edge_kernel<<<(E + 127) / 128, 256, 0, stream>>>(eattr, eemb, fcw1, fcw2, eidx,
                                                   s1, v1, S, V, E, N);
  node_out_kernel<<<(N + 127) / 128, 256, 0, stream>>>(nf, nattr, W2s, W2v, Wscs,
                                                       Wscv, S, V, out, N);
}